// TargetIdet_3745211482608
// MI455X (gfx1250) — compile-verified
//
#include <hip/hip_runtime.h>
#include <math.h>

#define EMB   128
#define HID   128
#define GATES 512      // 4*HID
#define NTOK  16
#define BATCH 4096
#define APAD  132      // padded LDS row stride (16B-aligned, conflict-friendly)

typedef __attribute__((ext_vector_type(2))) float v2f;
typedef __attribute__((ext_vector_type(8))) float v8f;

__device__ __forceinline__ v8f wmma_f32_16x16x4(v2f a, v2f b, v8f c) {
  // D = A(16x4 f32) * B(4x16 f32) + C(16x16 f32)
  return __builtin_amdgcn_wmma_f32_16x16x4_f32(
      /*neg_a=*/false, a, /*neg_b=*/false, b,
      /*c_mod=*/(short)0, c, /*reuse_a=*/false, /*reuse_b=*/false);
}

// ---------------------------------------------------------------------------
// Phase 1: zx[t, o] = emb[x[t]] . w_ih[o] + (b_ih[o] + b_hh[o])
// Grid 64 blocks x 128 threads. Each block owns 64 batch rows (4 M-tiles, one
// per wave); loops over the 32 N-tiles of the 512 gate outputs, staging the
// 16x128 w_ih tile and the gathered 64x128 embedding tile in LDS.
// ---------------------------------------------------------------------------
__global__ __launch_bounds__(128) void k_input_proj(
    const long long* __restrict__ x, const float* __restrict__ emb,
    const float* __restrict__ w_ih, const float* __restrict__ b_ih,
    const float* __restrict__ b_hh, float* __restrict__ zx)
{
  __shared__ float aS[64 * APAD];   // gathered embedding rows (4 M-tiles)
  __shared__ float bS[16 * APAD];   // one w_ih N-tile (K-major rows)

  const int tid  = threadIdx.x;
  const int lane = tid & 31;
  const int wave = tid >> 5;
  const int m0   = blockIdx.x * 64;

  // Gather 64 embedding rows into LDS (2 threads per row, float4 copies).
  {
    const int r    = tid >> 1;
    const int half = tid & 1;
    const long long tok = x[m0 + r];
    const float4* src = (const float4*)(emb + (size_t)tok * EMB + half * 64);
    float4*       dst = (float4*)(aS + r * APAD + half * 64);
#pragma unroll
    for (int i = 0; i < 16; ++i) dst[i] = src[i];
  }

  // WMMA fragment coordinates (ISA 7.12.2 layouts, wave32).
  const int m      = lane & 15;          // A row
  const int n      = lane & 15;          // B/C column
  const int kk     = (lane >> 4) * 2;    // K sub-pair per lane half
  const int rowsel = (lane >> 4) * 8;    // C row offset per lane half

  for (int nt = 0; nt < 32; ++nt) {
    __syncthreads();                      // aS ready / previous bS consumed
    {
      const int r = tid >> 3;             // 0..15 (row of tile)
      const int c = (tid & 7) * 16;       // 0..112
      const float4* src = (const float4*)(w_ih + (nt * 16 + r) * EMB + c);
      float4*       dst = (float4*)(bS + r * APAD + c);
#pragma unroll
      for (int i = 0; i < 4; ++i) dst[i] = src[i];
    }
    __syncthreads();

    v8f acc = {};
    const float* ap = aS + (wave * 16 + m) * APAD + kk;
    const float* bp = bS + n * APAD + kk;
#pragma unroll
    for (int k0 = 0; k0 < EMB; k0 += 4) {
      v2f a; a.x = ap[k0]; a.y = ap[k0 + 1];
      v2f b; b.x = bp[k0]; b.y = bp[k0 + 1];
      acc = wmma_f32_16x16x4(a, b, acc);
    }

    const int o    = nt * 16 + n;
    const float bi = b_ih[o] + b_hh[o];
#pragma unroll
    for (int v = 0; v < 8; ++v) {
      const int row = m0 + wave * 16 + rowsel + v;
      zx[row * GATES + o] = acc[v] + bi;
    }
  }
}

// ---------------------------------------------------------------------------
// Phase 2: sequential LSTM recurrence. One 512-thread workgroup; w_hh lives in
// LDS (512 x APAD floats = 264 KB, needs CDNA5's 320 KB LDS). Thread o owns
// gate pre-activation o; h/c/gates shuttle through LDS with split barriers.
// ---------------------------------------------------------------------------
__global__ __launch_bounds__(512) void k_lstm(
    const float* __restrict__ zx, const float* __restrict__ w_hh,
    float* __restrict__ hs)
{
  extern __shared__ float smem[];
  float* whh  = smem;                    // GATES * APAD
  float* hbuf = whh + GATES * APAD;      // HID
  float* cbuf = hbuf + HID;              // HID
  float* gbuf = cbuf + HID;              // GATES

  const int o = threadIdx.x;

  { // stage w_hh row o (padded stride keeps float4 loads 16B aligned)
    const float4* src = (const float4*)(w_hh + o * HID);
    float4*       dst = (float4*)(whh + o * APAD);
#pragma unroll
    for (int i = 0; i < HID / 4; ++i) dst[i] = src[i];
  }
  if (o < HID) { hbuf[o] = 0.0f; cbuf[o] = 0.0f; }
  __syncthreads();

  const int gate = o >> 7;               // 0=i 1=f 2=g 3=o
  const float4* wp = (const float4*)(whh + o * APAD);
  const float4* hp = (const float4*)hbuf;

  for (int t = 0; t < BATCH; ++t) {
    if (t + 4 < BATCH)                   // lowers to global_prefetch_b8
      __builtin_prefetch(&zx[(t + 4) * GATES + o], 0, 1);

    float acc = zx[t * GATES + o];
#pragma unroll 8
    for (int k = 0; k < HID / 4; ++k) {
      const float4 w4 = wp[k];
      const float4 h4 = hp[k];           // LDS broadcast across lanes
      acc = fmaf(w4.x, h4.x, acc);
      acc = fmaf(w4.y, h4.y, acc);
      acc = fmaf(w4.z, h4.z, acc);
      acc = fmaf(w4.w, h4.w, acc);
    }
    gbuf[o] = (gate == 2) ? tanhf(acc) : 1.0f / (1.0f + __expf(-acc));
    __syncthreads();                     // all gates visible

    if (o < HID) {
      const float c = gbuf[HID + o] * cbuf[o] + gbuf[o] * gbuf[2 * HID + o];
      cbuf[o] = c;
      const float h = gbuf[3 * HID + o] * tanhf(c);
      hbuf[o] = h;
      hs[t * HID + o] = h;
    }
    __syncthreads();                     // h/c updated before next matvec
  }
}

// ---------------------------------------------------------------------------
// Phase 3: out = hs @ w_fc.T + b_fc  (N = 16 == one WMMA tile column).
// Grid 64 x 128; each wave produces one 16x16 output tile.
// ---------------------------------------------------------------------------
__global__ __launch_bounds__(128) void k_fc(
    const float* __restrict__ hs, const float* __restrict__ w_fc,
    const float* __restrict__ b_fc, float* __restrict__ out)
{
  const int tid  = threadIdx.x;
  const int lane = tid & 31;
  const int wave = tid >> 5;
  const int mt   = blockIdx.x * 4 + wave;   // 0..255 M-tiles

  const int m      = lane & 15;
  const int n      = lane & 15;
  const int kk     = (lane >> 4) * 2;
  const int rowsel = (lane >> 4) * 8;

  const float* ap = hs + (mt * 16 + m) * HID + kk;
  const float* bp = w_fc + n * HID + kk;    // n < 16 == NTOK, always valid

  v8f acc = {};
#pragma unroll
  for (int k0 = 0; k0 < HID; k0 += 4) {
    v2f a; a.x = ap[k0]; a.y = ap[k0 + 1];
    v2f b; b.x = bp[k0]; b.y = bp[k0 + 1];
    acc = wmma_f32_16x16x4(a, b, acc);
  }

  const float bias = b_fc[n];
#pragma unroll
  for (int v = 0; v < 8; ++v) {
    const int row = mt * 16 + rowsel + v;
    out[row * NTOK + n] = acc[v] + bias;
  }
}

// ---------------------------------------------------------------------------
extern "C" void kernel_launch(void* const* d_in, const int* in_sizes, int n_in,
                              void* d_out, int out_size, void* d_ws, size_t ws_size,
                              hipStream_t stream) {
  const long long* x    = (const long long*)d_in[0];  // int64 tokens
  const float*     emb  = (const float*)d_in[1];
  const float*     w_ih = (const float*)d_in[2];
  const float*     w_hh = (const float*)d_in[3];
  const float*     b_ih = (const float*)d_in[4];
  const float*     b_hh = (const float*)d_in[5];
  const float*     w_fc = (const float*)d_in[6];
  const float*     b_fc = (const float*)d_in[7];
  float*           out  = (float*)d_out;

  float* zx = (float*)d_ws;               // BATCH*GATES f32 = 8 MB
  float* hs = zx + (size_t)BATCH * GATES; // BATCH*HID   f32 = 2 MB

  k_input_proj<<<64, 128, 0, stream>>>(x, emb, w_ih, b_ih, b_hh, zx);

  const size_t smem = (size_t)(GATES * APAD + HID + HID + GATES) * sizeof(float);
  k_lstm<<<1, 512, smem, stream>>>(zx, w_hh, hs);

  k_fc<<<64, 128, 0, stream>>>(hs, w_fc, b_fc, out);
}